// MultiBoxLoss_22162031247767
// MI455X (gfx1250) — compile-verified
//
#include <hip/hip_runtime.h>
#include <hip/hip_bf16.h>

#define NUM_B     256
#define NUM_P     8732
#define NUM_OBJ   32
#define NTHREADS  256
#define OV_THRESH 0.5f
#define VAR_0     0.1f
#define VAR_1     0.2f
#define NEGPOS    3

typedef __attribute__((ext_vector_type(2))) float v2f;
typedef __attribute__((ext_vector_type(8))) float v8f;

// Block-wide sum of 256 per-thread f32 partials using V_WMMA_F32_16X16X4_F32.
// Wave 0 (lanes 0..31, EXEC all ones) performs 4 accumulating WMMAs with an
// all-ones A (16x4) against B chunks (4x16) holding 64 partials each:
// D[m][n] = sum_k B[k][n], so row 0 of D carries 16 column sums covering all
// 256 values; a 16-element finish yields the exact total (pure f32 math).
// The result is identical on every thread (same 16 LDS reads, same order).
__device__ __forceinline__ float block_sum_wmma(float v, float* s_red) {
  const int tid = threadIdx.x;
  __syncthreads();            // protect prior consumers of s_red
  s_red[tid] = v;
  __syncthreads();
  if (tid < 32) {             // exactly wave 0 on wave32 hardware
    v2f A; A.x = 1.0f; A.y = 1.0f;   // all-ones 16x4 A operand
    v8f acc = {};
    const int n  = tid & 15;
    const int kh = (tid >> 4) << 1;
#pragma unroll
    for (int c = 0; c < 4; ++c) {
      const int cb = c << 6;
      v2f Bv;
      Bv.x = s_red[cb + kh * 16 + n];
      Bv.y = s_red[cb + (kh + 1) * 16 + n];
      acc = __builtin_amdgcn_wmma_f32_16x16x4_f32(false, A, false, Bv,
                                                  (short)0, acc, false, false);
    }
    if (tid < 16) s_red[tid] = acc[0];   // D[0][n], n = lane
  }
  __syncthreads();
  float t = 0.0f;
#pragma unroll
  for (int i = 0; i < 16; ++i) t += s_red[i];
  return t;
}

__global__ __launch_bounds__(NTHREADS) void multibox_main(
    const float* __restrict__ loc_data,    // [B,P,4]
    const float* __restrict__ conf_data,   // [B,P,2]
    const float* __restrict__ priors,      // [P,4] (cx,cy,w,h)
    const float* __restrict__ targets,     // [B,NOBJ,5]
    float* __restrict__ ws) {              // [3*B] partials

  __shared__ float         s_val[NUM_P];        // overlap, then masked-CE r[p]
  __shared__ unsigned char s_bj[NUM_P];         // best truth index per prior
  __shared__ float         s_truth[NUM_OBJ][6]; // x1,y1,x2,y2,area,label
  __shared__ float         s_red[NTHREADS];     // WMMA reduction scratch
  __shared__ float         s_redv[NTHREADS];    // argmax reduce (value)
  __shared__ int           s_redi[NTHREADS];    // argmax reduce (index)
  __shared__ int           s_bpi[NUM_OBJ];      // best prior per truth

  const int tid = threadIdx.x;
  const int b   = blockIdx.x;

  if (tid < NUM_OBJ) {
    const float* t = targets + ((size_t)b * NUM_OBJ + tid) * 5;
    const float x1 = t[0], y1 = t[1], x2 = t[2], y2 = t[3];
    s_truth[tid][0] = x1; s_truth[tid][1] = y1;
    s_truth[tid][2] = x2; s_truth[tid][3] = y2;
    s_truth[tid][4] = (x2 - x1) * (y2 - y1);
    s_truth[tid][5] = t[4];
  }
  __syncthreads();

  // ---- Phase 1: IoU pass. Per-prior best truth -> LDS; per-truth best prior
  // kept per-thread in registers (32 unrolled slots), reduced afterwards.
  float tbv[NUM_OBJ];
  int   tbp[NUM_OBJ];
#pragma unroll
  for (int j = 0; j < NUM_OBJ; ++j) { tbv[j] = -1.0f; tbp[j] = NUM_P; }

  for (int p = tid; p < NUM_P; p += NTHREADS) {
    const float4 pr  = *reinterpret_cast<const float4*>(priors + (size_t)p * 4);
    const float px1  = pr.x - pr.z * 0.5f;
    const float py1  = pr.y - pr.w * 0.5f;
    const float px2  = pr.x + pr.z * 0.5f;
    const float py2  = pr.y + pr.w * 0.5f;
    const float pa   = (px2 - px1) * (py2 - py1);   // area from point form (matches ref)
    float bestv = -1.0f; int bestj = 0;
#pragma unroll
    for (int j = 0; j < NUM_OBJ; ++j) {
      const float ix1 = fmaxf(s_truth[j][0], px1);
      const float iy1 = fmaxf(s_truth[j][1], py1);
      const float ix2 = fminf(s_truth[j][2], px2);
      const float iy2 = fminf(s_truth[j][3], py2);
      const float iw  = fmaxf(ix2 - ix1, 0.0f);
      const float ih  = fmaxf(iy2 - iy1, 0.0f);
      const float inter = iw * ih;
      const float iou   = inter / (s_truth[j][4] + pa - inter);
      if (iou > tbv[j]) { tbv[j] = iou; tbp[j] = p; }   // first-index on ties
      if (iou > bestv)  { bestv = iou; bestj = j; }     // first-index on ties
    }
    s_val[p] = bestv;
    s_bj[p]  = (unsigned char)bestj;
  }
  __syncthreads();

  // ---- Phase 1b: reduce per-truth argmax across the 256 threads.
  for (int j = 0; j < NUM_OBJ; ++j) {
    s_redv[tid] = tbv[j];
    s_redi[tid] = tbp[j];
    __syncthreads();
    for (int s = NTHREADS / 2; s > 0; s >>= 1) {
      if (tid < s) {
        const float v2 = s_redv[tid + s]; const int p2 = s_redi[tid + s];
        if (v2 > s_redv[tid] || (v2 == s_redv[tid] && p2 < s_redi[tid])) {
          s_redv[tid] = v2; s_redi[tid] = p2;
        }
      }
      __syncthreads();
    }
    if (tid == 0) s_bpi[j] = s_redi[0];
    __syncthreads();
  }

  // ---- Phase 2: force each truth's best prior (sequential: later truth wins,
  // matching scatter-with-duplicates semantics).
  if (tid == 0) {
    for (int j = 0; j < NUM_OBJ; ++j) {
      const int bp = s_bpi[j];
      s_val[bp] = 2.0f;
      s_bj[bp]  = (unsigned char)j;
    }
  }
  __syncthreads();

  // ---- Phase 3: smooth-L1 over positives, CE per prior, mining values r[p].
  float lossL = 0.0f, cePos = 0.0f, npos = 0.0f;
  for (int p = tid; p < NUM_P; p += NTHREADS) {
    const float ov  = s_val[p];
    const int   j   = (int)s_bj[p];
    const bool  pos = (ov >= OV_THRESH);
    const float2 cf = *reinterpret_cast<const float2*>(conf_data + ((size_t)b * NUM_P + p) * 2);
    const float m   = fmaxf(cf.x, cf.y);
    const float lse = m + logf(expf(cf.x - m) + expf(cf.y - m));
    if (pos) {
      const int   cls      = (int)s_truth[j][5] + 1;     // label 0 -> class 1
      const float gathered = (cls == 0) ? cf.x : cf.y;
      cePos += lse - gathered;
      npos  += 1.0f;
      const float4 pr = *reinterpret_cast<const float4*>(priors + (size_t)p * 4);
      const float4 ld = *reinterpret_cast<const float4*>(loc_data + ((size_t)b * NUM_P + p) * 4);
      const float mx1 = s_truth[j][0], my1 = s_truth[j][1];
      const float mx2 = s_truth[j][2], my2 = s_truth[j][3];
      const float gx = ((mx1 + mx2) * 0.5f - pr.x) / (VAR_0 * pr.z);
      const float gy = ((my1 + my2) * 0.5f - pr.y) / (VAR_0 * pr.w);
      const float gw = logf((mx2 - mx1) / pr.z) / VAR_1;
      const float gh = logf((my2 - my1) / pr.w) / VAR_1;
      float d, ad;
      d = ld.x - gx; ad = fabsf(d); lossL += (ad < 1.0f) ? 0.5f * d * d : ad - 0.5f;
      d = ld.y - gy; ad = fabsf(d); lossL += (ad < 1.0f) ? 0.5f * d * d : ad - 0.5f;
      d = ld.z - gw; ad = fabsf(d); lossL += (ad < 1.0f) ? 0.5f * d * d : ad - 0.5f;
      d = ld.w - gh; ad = fabsf(d); lossL += (ad < 1.0f) ? 0.5f * d * d : ad - 0.5f;
      s_val[p] = 0.0f;            // positives ranked as 0
    } else {
      s_val[p] = lse - cf.x;      // CE against class 0 (>= 0)
    }
  }

  const float numPos   = block_sum_wmma(npos,  s_red);
  const float sumCePos = block_sum_wmma(cePos, s_red);
  const float sumLossL = block_sum_wmma(lossL, s_red);

  int k = NEGPOS * (int)(numPos + 0.5f);
  if (k > NUM_P - 1) k = NUM_P - 1;

  // ---- Phase 4: top-k sum via bitwise radix-select (r >= 0 so the float's
  // uint image is order-preserving). Sum = sum(r>T) + (k - count(r>T)) * T,
  // exact under ties, so no sort is needed.
  float negSum = 0.0f;
  if (k > 0) {
    unsigned prefix = 0u;
    for (int bit = 31; bit >= 0; --bit) {
      const unsigned X = prefix | (1u << bit);
      float cnt = 0.0f;
      for (int p = tid; p < NUM_P; p += NTHREADS)
        cnt += (__float_as_uint(s_val[p]) >= X) ? 1.0f : 0.0f;
      const float total = block_sum_wmma(cnt, s_red);   // exact int in f32
      if (total >= (float)k) prefix = X;                // uniform decision
    }
    const float T = __uint_as_float(prefix);            // k-th largest value
    float sgt = 0.0f, cgt = 0.0f;
    for (int p = tid; p < NUM_P; p += NTHREADS) {
      const float r = s_val[p];
      if (r > T) { sgt += r; cgt += 1.0f; }
    }
    const float SGT = block_sum_wmma(sgt, s_red);
    const float CGT = block_sum_wmma(cgt, s_red);
    negSum = SGT + ((float)k - CGT) * T;
  }

  if (tid == 0) {
    ws[b]             = sumLossL;
    ws[NUM_B + b]     = sumCePos + negSum;
    ws[2 * NUM_B + b] = numPos;
  }
}

// Deterministic final reduction (fixed summation order; no float atomics).
__global__ void multibox_finalize(const float* __restrict__ ws,
                                  float* __restrict__ out) {
  float sL = 0.0f, sC = 0.0f, sN = 0.0f;
  for (int i = 0; i < NUM_B; ++i) {
    sL += ws[i];
    sC += ws[NUM_B + i];
    sN += ws[2 * NUM_B + i];
  }
  out[0] = sL / sN;
  out[1] = sC / sN;
}

extern "C" void kernel_launch(void* const* d_in, const int* in_sizes, int n_in,
                              void* d_out, int out_size, void* d_ws, size_t ws_size,
                              hipStream_t stream) {
  (void)in_sizes; (void)n_in; (void)out_size; (void)ws_size;
  const float* loc     = (const float*)d_in[0];
  const float* conf    = (const float*)d_in[1];
  const float* priors  = (const float*)d_in[2];
  const float* targets = (const float*)d_in[3];
  float* ws  = (float*)d_ws;
  float* out = (float*)d_out;
  multibox_main<<<dim3(NUM_B), dim3(NTHREADS), 0, stream>>>(loc, conf, priors, targets, ws);
  multibox_finalize<<<1, 1, 0, stream>>>(ws, out);
}